// SimpleGCN_62886911148524
// MI455X (gfx1250) — compile-verified
//
#include <hip/hip_runtime.h>
#include <hip/hip_bf16.h>

typedef __attribute__((ext_vector_type(2))) float v2f;
typedef __attribute__((ext_vector_type(8))) float v8f;

#define C_DIM 32

// ---------------------------------------------------------------------------
// Degree kernels: deg[i] = 1 (self loop) + #edges with col==i; then rsqrt.
// ---------------------------------------------------------------------------
__global__ void init_deg_kernel(float* __restrict__ deg, int n_nodes) {
    int i = blockIdx.x * blockDim.x + threadIdx.x;
    if (i < n_nodes) deg[i] = 1.0f;  // self-loop contribution
}

__global__ void count_deg_kernel(const int* __restrict__ col, float* __restrict__ deg,
                                 int n_edges) {
    int e = blockIdx.x * blockDim.x + threadIdx.x;
    if (e < n_edges) atomicAdd(&deg[col[e]], 1.0f);
}

__global__ void rsqrt_kernel(float* __restrict__ deg, int n_nodes) {
    int i = blockIdx.x * blockDim.x + threadIdx.x;
    if (i < n_nodes) {
        float d = deg[i];
        deg[i] = (d > 0.0f) ? __frsqrt_rn(d) : 0.0f;  // deg -> dis in place
    }
}

// ---------------------------------------------------------------------------
// WMMA GEMM: H[n,32] = A[n,32] @ W[32,32], fp32 via V_WMMA_F32_16X16X4_F32.
// One wave computes a 16-row x 32-col strip: 2 accumulators x 8 K-chunks.
// MODE 0: plain load.  MODE 2: A := relu(A) on load (layer-2 input; bias was
// already folded into the aggregation buffer by init_agg_kernel).
//
// f32 16x4 A fragment layout (ISA 7.12.2): lanes 0-15 -> (M=lane, K=0/1),
// lanes 16-31 -> (M=lane-16, K=2/3). B 4x16 mirrors with N across lanes.
// C/D 16x16: VGPR r, lanes 0-15 -> M=r, lanes 16-31 -> M=r+8, N=lane&15.
// ---------------------------------------------------------------------------
template <int MODE>
__global__ void gemm_n32_kernel(const float* __restrict__ A,
                                const float* __restrict__ W,
                                float* __restrict__ H, int n_nodes) {
    const int wave = threadIdx.x >> 5;
    const int lane = threadIdx.x & 31;
    const int tile = blockIdx.x * (blockDim.x >> 5) + wave;  // 16-row tile
    const int row0 = tile * 16;
    if (row0 >= n_nodes) return;

    const int m  = lane & 15;          // M (for A) / N (for B) within half-wave
    const int kh = (lane >> 4) << 1;   // 0 for lanes 0-15, 2 for lanes 16-31
    const int ar = (row0 + m < n_nodes) ? (row0 + m) : (n_nodes - 1);

    v8f c0 = {};
    v8f c1 = {};

#pragma unroll
    for (int kc = 0; kc < 8; ++kc) {
        const int k = kc * 4 + kh;
        float a0 = A[ar * C_DIM + k];
        float a1 = A[ar * C_DIM + k + 1];
        if (MODE == 2) {  // relu on load (compile-time branch)
            a0 = fmaxf(a0, 0.0f);
            a1 = fmaxf(a1, 0.0f);
        }
        v2f av = {a0, a1};
        v2f b0 = {W[k * C_DIM + m],      W[(k + 1) * C_DIM + m]};
        v2f b1 = {W[k * C_DIM + 16 + m], W[(k + 1) * C_DIM + 16 + m]};
        c0 = __builtin_amdgcn_wmma_f32_16x16x4_f32(false, av, false, b0,
                                                   (short)0, c0, false, false);
        c1 = __builtin_amdgcn_wmma_f32_16x16x4_f32(false, av, false, b1,
                                                   (short)0, c1, false, false);
    }

    const int moff = (lane >> 4) << 3;  // rows r (lanes 0-15) / r+8 (lanes 16-31)
#pragma unroll
    for (int r = 0; r < 8; ++r) {
        const int row = row0 + r + moff;
        if (row < n_nodes) {
            H[row * C_DIM + m]      = c0[r];
            H[row * C_DIM + 16 + m] = c1[r];
        }
    }
}

// ---------------------------------------------------------------------------
// agg[i,c] = dis[i]^2 * h[i,c] + bias[c]   (self-loop term + output bias)
// thread = (node, channel); lane = channel -> coalesced 128B rows.
// ---------------------------------------------------------------------------
__global__ void init_agg_kernel(const float* __restrict__ h,
                                const float* __restrict__ dis,
                                const float* __restrict__ bias,
                                float* __restrict__ agg, int n_nodes) {
    int idx = blockIdx.x * blockDim.x + threadIdx.x;
    int i = idx >> 5;
    int c = idx & 31;
    if (i < n_nodes) {
        float d = dis[i];
        agg[i * C_DIM + c] = d * d * h[i * C_DIM + c] + bias[c];
    }
}

// ---------------------------------------------------------------------------
// Edge scatter: one wave per edge, lane = channel.
// agg[col,c] += dis[row]*dis[col] * h[row,c]  via global_atomic_add_f32.
// ---------------------------------------------------------------------------
__global__ void scatter_edges_kernel(const float* __restrict__ h,
                                     const float* __restrict__ dis,
                                     const int* __restrict__ row,
                                     const int* __restrict__ col,
                                     float* __restrict__ agg, int n_edges) {
    int idx = blockIdx.x * blockDim.x + threadIdx.x;
    int e = idx >> 5;
    int c = idx & 31;
    if (e < n_edges) {
        int r  = row[e];
        int cl = col[e];
        float nrm = dis[r] * dis[cl];
        atomicAdd(&agg[cl * C_DIM + c], nrm * h[r * C_DIM + c]);
    }
}

// ---------------------------------------------------------------------------
// Launch
// ---------------------------------------------------------------------------
extern "C" void kernel_launch(void* const* d_in, const int* in_sizes, int n_in,
                              void* d_out, int out_size, void* d_ws, size_t ws_size,
                              hipStream_t stream) {
    const float* x    = (const float*)d_in[0];
    const int*   eidx = (const int*)d_in[1];   // [2, E] flat: row then col
    const float* W1   = (const float*)d_in[2];
    const float* b1   = (const float*)d_in[3];
    const float* W2   = (const float*)d_in[4];
    const float* b2   = (const float*)d_in[5];

    const int n_nodes = in_sizes[0] / C_DIM;
    const int n_edges = in_sizes[1] / 2;
    const int* row = eidx;
    const int* col = eidx + n_edges;

    // workspace layout: dis[N] | h[N*32] | agg1[N*32]   (~26 MB)
    float* dis  = (float*)d_ws;
    float* h    = dis + n_nodes;
    float* agg1 = h + (size_t)n_nodes * C_DIM;
    float* out  = (float*)d_out;

    const int T = 256;
    const int waves_per_blk = T / 32;
    const int n_blk  = (n_nodes + T - 1) / T;
    const int e_blk  = (n_edges + T - 1) / T;
    const int nc_blk = (int)(((size_t)n_nodes * 32 + T - 1) / T);
    const int ec_blk = (int)(((size_t)n_edges * 32 + T - 1) / T);
    const int tiles  = (n_nodes + 15) / 16;
    const int g_blk  = (tiles + waves_per_blk - 1) / waves_per_blk;

    // degrees -> dis (in place)
    init_deg_kernel<<<n_blk, T, 0, stream>>>(dis, n_nodes);
    count_deg_kernel<<<e_blk, T, 0, stream>>>(col, dis, n_edges);
    rsqrt_kernel<<<n_blk, T, 0, stream>>>(dis, n_nodes);

    // ----- layer 1 -----  h = x @ W1 ; agg1 = scatter(norm * h) + b1
    gemm_n32_kernel<0><<<g_blk, T, 0, stream>>>(x, W1, h, n_nodes);
    init_agg_kernel<<<nc_blk, T, 0, stream>>>(h, dis, b1, agg1, n_nodes);
    scatter_edges_kernel<<<ec_blk, T, 0, stream>>>(h, dis, row, col, agg1, n_edges);

    // ----- layer 2 -----  h = relu(agg1) @ W2 ; out = scatter(norm * h) + b2
    gemm_n32_kernel<2><<<g_blk, T, 0, stream>>>(agg1, W2, h, n_nodes);
    init_agg_kernel<<<nc_blk, T, 0, stream>>>(h, dis, b2, out, n_nodes);
    scatter_edges_kernel<<<ec_blk, T, 0, stream>>>(h, dis, row, col, out, n_edges);
}